// SimAttention_60301340835943
// MI455X (gfx1250) — compile-verified
//
#include <hip/hip_runtime.h>

typedef __attribute__((ext_vector_type(2))) float v2f;
typedef __attribute__((ext_vector_type(8))) float v8f;

#define B_ 64
#define N_ 8192
#define D_ 128
#define THRESH 0.9f
#define EPS_ 1e-8f

// ---------------- Kernel 0: per-batch query norm + zero accumulators ----------------
__global__ void qnorm_init_kernel(const float* __restrict__ q,
                                  float* __restrict__ qn,
                                  float* __restrict__ sums,
                                  float* __restrict__ acc) {
  const int b = blockIdx.x;
  const int t = threadIdx.x;
  float v = q[b * D_ + t];
  __shared__ float r[D_];
  r[t] = v * v;
  __syncthreads();
  #pragma unroll
  for (int s = D_ / 2; s > 0; s >>= 1) {
    if (t < s) r[t] += r[t + s];
    __syncthreads();
  }
  if (t == 0) {
    qn[b] = sqrtf(r[0]);
    sums[b] = 0.0f;
  }
  acc[b * D_ + t] = 0.0f;
}

// ---------------- Kernel 1: WMMA dot products + cosine + threshold ----------------
// One wave computes dot(key[b, n0..n0+15, :], q[b]) via 32 chained
// v_wmma_f32_16x16x4_f32 ops. B operand = query broadcast to all 16 columns,
// so the A operand (the streamed key data) is used at full memory efficiency.
__global__ void attn_kernel(const float* __restrict__ query,
                            const float* __restrict__ key,
                            const unsigned char* __restrict__ mask,
                            const float* __restrict__ qn,
                            float* __restrict__ attn,
                            float* __restrict__ sums) {
  const int b    = blockIdx.y;
  const int wave = threadIdx.x >> 5;
  const int lane = threadIdx.x & 31;
  const int half = lane >> 4;      // 0: K%4 in {0,1}, 1: K%4 in {2,3}
  const int row  = lane & 15;      // A-matrix row (n within tile)
  const int n0   = blockIdx.x * 128 + wave * 16;

  const float* krow = key + ((size_t)b * N_ + (size_t)(n0 + row)) * D_;
  const float* qv   = query + (size_t)b * D_;

  v8f c0 = {};
  v8f c1 = {};
  float nk2p = 0.0f;

  #pragma unroll
  for (int k0 = 0; k0 < D_; k0 += 8) {
    const int d0 = k0 + 2 * half;
    v2f a0 = *(const v2f*)(krow + d0);
    v2f b0 = *(const v2f*)(qv + d0);
    c0 = __builtin_amdgcn_wmma_f32_16x16x4_f32(false, a0, false, b0,
                                               (short)0, c0, false, false);
    nk2p = fmaf(a0.x, a0.x, fmaf(a0.y, a0.y, nk2p));

    const int d1 = d0 + 4;
    v2f a1 = *(const v2f*)(krow + d1);
    v2f b1 = *(const v2f*)(qv + d1);
    c1 = __builtin_amdgcn_wmma_f32_16x16x4_f32(false, a1, false, b1,
                                               (short)0, c1, false, false);
    nk2p = fmaf(a1.x, a1.x, fmaf(a1.y, a1.y, nk2p));
  }
  v8f c = c0 + c1;

  // ||key_row||^2: lane l and l^16 hold the two K%4 halves of row (l&15)
  float nk2 = nk2p + __shfl_xor(nk2p, 16);

  // D layout: VGPR r on lane l holds row r + 8*(l>=16); columns identical.
  // Lane l extracts row m = (l&7) + 8*half via a select chain.
  const int rsel = lane & 7;
  float dotv = c[0];
  if (rsel == 1) dotv = c[1];
  if (rsel == 2) dotv = c[2];
  if (rsel == 3) dotv = c[3];
  if (rsel == 4) dotv = c[4];
  if (rsel == 5) dotv = c[5];
  if (rsel == 6) dotv = c[6];
  if (rsel == 7) dotv = c[7];
  const int m = rsel + 8 * half;

  // ||key||^2 for row m lives on lane m
  float nk2m = __shfl(nk2, m);

  const float nq    = qn[b];
  const float denom = fmaxf(nq * sqrtf(nk2m), EPS_);
  float av = dotv / denom;

  const int n = n0 + m;
  if (mask[(size_t)b * N_ + n]) av = 0.0f;
  if (av < THRESH) av = 0.0f;

  // rows 0..15 are each produced by two lanes; keep one producer per row
  const bool act = (lane & 15) < 8;
  if (act) attn[(size_t)b * N_ + n] = av;

  float s = act ? av : 0.0f;
  s += __shfl_xor(s, 1);
  s += __shfl_xor(s, 2);
  s += __shfl_xor(s, 4);
  s += __shfl_xor(s, 8);
  s += __shfl_xor(s, 16);
  if (lane == 0) atomicAdd(&sums[b], s);
}

// ---------------- Kernel 2: weighted context (skips zero-weight rows) ----------------
__global__ void wsum_kernel(const float* __restrict__ value,
                            const float* __restrict__ attn,
                            float* __restrict__ acc) {
  const int b  = blockIdx.y;
  const int n0 = blockIdx.x * 256;
  const int t  = threadIdx.x;
  const int dg = (t & 31) * 4;   // 4 consecutive d per lane -> 512B/wave loads
  const int nl = t >> 5;         // 8 n-lanes (one wave each)

  float4 a = make_float4(0.f, 0.f, 0.f, 0.f);
  for (int n = n0 + nl; n < n0 + 256; n += 8) {
    const float w = attn[(size_t)b * N_ + n];
    if (w != 0.0f) {  // wave-uniform: whole value row load skipped when weight==0
      const float4 v = *(const float4*)(value + ((size_t)b * N_ + n) * D_ + dg);
      a.x = fmaf(w, v.x, a.x);
      a.y = fmaf(w, v.y, a.y);
      a.z = fmaf(w, v.z, a.z);
      a.w = fmaf(w, v.w, a.w);
    }
  }

  __shared__ float4 red[256];
  red[t] = a;
  __syncthreads();
  if (nl == 0) {
    #pragma unroll
    for (int j = 1; j < 8; ++j) {
      const float4 o = red[t + 32 * j];
      a.x += o.x; a.y += o.y; a.z += o.z; a.w += o.w;
    }
    atomicAdd(&acc[b * D_ + dg + 0], a.x);
    atomicAdd(&acc[b * D_ + dg + 1], a.y);
    atomicAdd(&acc[b * D_ + dg + 2], a.z);
    atomicAdd(&acc[b * D_ + dg + 3], a.w);
  }
}

// ---------------- Kernel 3: normalize by (sum + eps) ----------------
__global__ void finalize_kernel(const float* __restrict__ acc,
                                const float* __restrict__ sums,
                                float* __restrict__ out) {
  const int b = blockIdx.x;
  const int t = threadIdx.x;
  out[b * D_ + t] = acc[b * D_ + t] / (sums[b] + EPS_);
}

extern "C" void kernel_launch(void* const* d_in, const int* in_sizes, int n_in,
                              void* d_out, int out_size, void* d_ws, size_t ws_size,
                              hipStream_t stream) {
  const float* query = (const float*)d_in[0];
  const float* key   = (const float*)d_in[1];
  const float* value = (const float*)d_in[2];
  const unsigned char* mask = (const unsigned char*)d_in[3];
  float* out = (float*)d_out;

  // workspace layout (~2.1 MB)
  float* attn = (float*)d_ws;                 // B*N
  float* sums = attn + (size_t)B_ * N_;       // B
  float* qn   = sums + B_;                    // B
  float* acc  = qn + B_;                      // B*D

  qnorm_init_kernel<<<dim3(B_), dim3(D_), 0, stream>>>(query, qn, sums, acc);
  attn_kernel<<<dim3(N_ / 128, B_), dim3(256), 0, stream>>>(query, key, mask, qn,
                                                            attn, sums);
  wsum_kernel<<<dim3(N_ / 256, B_), dim3(256), 0, stream>>>(value, attn, acc);
  finalize_kernel<<<dim3(B_), dim3(D_), 0, stream>>>(acc, sums, out);
}